// GraphAttentionLayer_12987981103701
// MI455X (gfx1250) — compile-verified
//
#include <hip/hip_runtime.h>
#include <hip/hip_bf16.h>
#include <math.h>

#define N_NODES 50000
#define N_EDGES 800000
#define IN_DIM  128
#define HD      128      // NUM_HEADS * OUT_DIM
#define NUM_HEADS 8
#define OUT_DIM 16

typedef __attribute__((ext_vector_type(2))) float v2f;
typedef __attribute__((ext_vector_type(8))) float v8f;

// D = A(16x4 f32) * B(4x16 f32) + C(16x16 f32), wave32 WMMA
__device__ __forceinline__ v8f wmma_f32_16x16x4(v2f a, v2f b, v8f c) {
  return __builtin_amdgcn_wmma_f32_16x16x4_f32(
      /*neg_a=*/false, a, /*neg_b=*/false, b,
      /*c_mod=*/(short)0, c, /*reuse_a=*/false, /*reuse_b=*/false);
}

// Computes one 16x16 output tile of  A(16 x 128) @ W(128 x 128)[:, colbase:colbase+16]
// A rows are contiguous (lda = 128), W is row-major 128x128.
// Per ISA layout for V_WMMA_F32_16X16X4_F32:
//   A: lanes 0-15 hold M=lane, K={4k,4k+1}; lanes 16-31 hold M=lane-16, K={4k+2,4k+3}
//   B: lanes 0-15 hold N=lane, K={4k,4k+1}; lanes 16-31 hold N=lane-16, K={4k+2,4k+3}
__device__ __forceinline__ v8f gemm_tile_16x16x128(const float* __restrict__ Arow,
                                                   const float* __restrict__ W,
                                                   int colbase) {
  const int lane = threadIdx.x & 31;
  const int half = lane >> 4;     // 0 or 1
  const int l15  = lane & 15;
  v8f c = {};
  const float* aptr = Arow + (size_t)l15 * IN_DIM + 2 * half;
  const float* bptr = W + colbase + l15 + (size_t)(2 * half) * HD;
  __builtin_prefetch(bptr, 0, 0);   // global_prefetch_b8 on the weight tile
#pragma unroll
  for (int k = 0; k < 32; ++k) {
    v2f a = *(const v2f*)(aptr + 4 * k);
    v2f b;
    b.x = bptr[(size_t)(4 * k) * HD];
    b.y = bptr[(size_t)(4 * k) * HD + HD];
    c = wmma_f32_16x16x4(a, b, c);
  }
  return c;
}

// ---------------- Kernel 1: Q/K/V node projections -----------------------
// grid = (N_NODES/16, 3), block = 256 (8 waves). Wave w -> col tile w.
__global__ void gt_proj_nodes(const float* __restrict__ h,
                              const float* __restrict__ Wq, const float* __restrict__ bq,
                              const float* __restrict__ Wk, const float* __restrict__ bk,
                              const float* __restrict__ Wv, const float* __restrict__ bv,
                              float* __restrict__ Q, float* __restrict__ K,
                              float* __restrict__ V) {
  const int n0 = blockIdx.x * 16;
  const int m  = blockIdx.y;                 // 0=Q,1=K,2=V
  const float* W    = (m == 0) ? Wq : ((m == 1) ? Wk : Wv);
  const float* bias = (m == 0) ? bq : ((m == 1) ? bk : bv);
  float* out        = (m == 0) ? Q  : ((m == 1) ? K  : V);

  const int colbase = (threadIdx.x >> 5) * 16;
  v8f c = gemm_tile_16x16x128(h + (size_t)n0 * IN_DIM, W, colbase);

  const int lane = threadIdx.x & 31;
  const int half = lane >> 4, l15 = lane & 15;
  const float bcol = bias[colbase + l15];
#pragma unroll
  for (int r = 0; r < 8; ++r)
    out[(size_t)(n0 + r + 8 * half) * HD + colbase + l15] = c[r] + bcol;
}

// ---------------- Kernel 2: per-edge-tile fused kernel -------------------
// grid = N_EDGES/16, block = 256 (8 waves).
__global__ void gt_edge_kernel(const float* __restrict__ e,
                               const int* __restrict__ src, const int* __restrict__ dst,
                               const float* __restrict__ We, const float* __restrict__ be,
                               const float* __restrict__ Q, const float* __restrict__ K,
                               const float* __restrict__ V,
                               float* __restrict__ wV, float* __restrict__ z,
                               float* __restrict__ e_out) {
  __shared__ __align__(16) float sK[16][HD];
  __shared__ __align__(16) float sQ[16][HD];
  __shared__ __align__(16) float sV[16][HD];
  __shared__ __align__(16) float sS[16][HD];   // Ep, then score (in place)
  __shared__ float ss[16][NUM_HEADS];
  __shared__ int   sDst[16];

  const int t  = threadIdx.x;
  const int e0 = blockIdx.x * 16;

  // Phase 0: gather K[src], Q[dst], V[src] for the 16 edges (8 floats/thread each)
  {
    const int i  = t >> 4;
    const int c0 = (t & 15) << 3;
    const int si = src[e0 + i];
    const int di = dst[e0 + i];
    if (t < 16) sDst[t] = dst[e0 + t];
    const float4* kp = (const float4*)(K + (size_t)si * HD + c0);
    const float4* qp = (const float4*)(Q + (size_t)di * HD + c0);
    const float4* vp = (const float4*)(V + (size_t)si * HD + c0);
    *(float4*)&sK[i][c0] = kp[0];  *(float4*)&sK[i][c0 + 4] = kp[1];
    *(float4*)&sQ[i][c0] = qp[0];  *(float4*)&sQ[i][c0 + 4] = qp[1];
    *(float4*)&sV[i][c0] = vp[0];  *(float4*)&sV[i][c0 + 4] = vp[1];
  }

  // Phase 1: Ep tile = e[e0:e0+16] @ We via WMMA, into sS
  {
    const int colbase = (t >> 5) * 16;
    v8f c = gemm_tile_16x16x128(e + (size_t)e0 * IN_DIM, We, colbase);
    const int lane = t & 31;
    const int half = lane >> 4, l15 = lane & 15;
    const float bcol = be[colbase + l15];
#pragma unroll
    for (int r = 0; r < 8; ++r)
      sS[r + 8 * half][colbase + l15] = c[r] + bcol;
  }
  __syncthreads();

  // Phase 2: score = K_src * Q_dst / sqrt(D) * Ep ; write e_out, keep in sS
  const float invScale = 0.25f;                  // 1/sqrt(16)
#pragma unroll
  for (int rep = 0; rep < 8; ++rep) {
    const int idx = rep * 256 + t;
    const int i = idx >> 7, cc = idx & 127;
    const float sc = sK[i][cc] * sQ[i][cc] * invScale * sS[i][cc];
    sS[i][cc] = sc;
    e_out[(size_t)(e0 + i) * HD + cc] = sc;
  }
  __syncthreads();

  // Phase 3: per (edge, head) attention weight s = exp(clamp(sum_d score))
  if (t < 128) {
    const int i = t >> 3, hd = t & 7;
    float sum = 0.f;
#pragma unroll
    for (int d = 0; d < OUT_DIM; ++d) sum += sS[i][hd * OUT_DIM + d];
    sum = fminf(5.0f, fmaxf(-5.0f, sum));
    ss[i][hd] = expf(sum);
  }
  __syncthreads();

  // Phase 4: scatter-add  wV[dst] += V_src * s ,  z[dst] += s
#pragma unroll
  for (int rep = 0; rep < 8; ++rep) {
    const int idx = rep * 256 + t;
    const int i = idx >> 7, cc = idx & 127;
    atomicAdd(&wV[(size_t)sDst[i] * HD + cc], sV[i][cc] * ss[i][cc >> 4]);
  }
  if (t < 128) {
    const int i = t >> 3, hd = t & 7;
    atomicAdd(&z[(size_t)sDst[i] * NUM_HEADS + hd], ss[i][hd]);
  }
}

// ---------------- Kernel 3: h_out = wV / (z + 1e-6) ----------------------
__global__ void gt_finalize(const float* __restrict__ wV, const float* __restrict__ z,
                            float* __restrict__ h_out) {
  const int idx = blockIdx.x * 256 + threadIdx.x;
  if (idx < N_NODES * HD) {
    const int n = idx >> 7, cc = idx & 127;
    h_out[idx] = wV[idx] / (z[n * NUM_HEADS + (cc >> 4)] + 1e-6f);
  }
}

extern "C" void kernel_launch(void* const* d_in, const int* in_sizes, int n_in,
                              void* d_out, int out_size, void* d_ws, size_t ws_size,
                              hipStream_t stream) {
  const float* h   = (const float*)d_in[0];
  const float* e   = (const float*)d_in[1];
  const int*   src = (const int*)  d_in[2];
  const int*   dst = (const int*)  d_in[3];
  const float* Wq  = (const float*)d_in[4];  const float* bq = (const float*)d_in[5];
  const float* Wk  = (const float*)d_in[6];  const float* bk = (const float*)d_in[7];
  const float* Wv  = (const float*)d_in[8];  const float* bv = (const float*)d_in[9];
  const float* We  = (const float*)d_in[10]; const float* be = (const float*)d_in[11];

  float* out   = (float*)d_out;
  float* h_out = out;                                  // [N_NODES, 128]
  float* e_out = out + (size_t)N_NODES * HD;           // [N_EDGES, 128]

  float* ws = (float*)d_ws;
  float* Q  = ws;                                      // N*128
  float* K  = ws + (size_t)N_NODES * HD;               // N*128
  float* V  = ws + (size_t)2 * N_NODES * HD;           // N*128
  float* wV = ws + (size_t)3 * N_NODES * HD;           // N*128 (accumulator)
  float* z  = ws + (size_t)4 * N_NODES * HD;           // N*8   (accumulator)

  // zero the scatter accumulators every call (harness does not re-poison)
  hipMemsetAsync(wV, 0,
                 ((size_t)N_NODES * HD + (size_t)N_NODES * NUM_HEADS) * sizeof(float),
                 stream);

  gt_proj_nodes<<<dim3(N_NODES / 16, 3), 256, 0, stream>>>(
      h, Wq, bq, Wk, bk, Wv, bv, Q, K, V);

  gt_edge_kernel<<<N_EDGES / 16, 256, 0, stream>>>(
      e, src, dst, We, be, Q, K, V, wV, z, e_out);

  gt_finalize<<<(N_NODES * HD + 255) / 256, 256, 0, stream>>>(wV, z, h_out);
}